// MemoryReader_23845658428024
// MI455X (gfx1250) — compile-verified
//
#include <hip/hip_runtime.h>
#include <hip/hip_bf16.h>
#include <float.h>
#include <math.h>

// Problem constants (from reference)
#define B_    16
#define H_    8
#define M_    64
#define K_    32
#define R_    65536
#define DIN_  512
#define OUTD_ 520   // (M+1)*H

typedef __attribute__((ext_vector_type(2))) float v2f;
typedef __attribute__((ext_vector_type(8))) float v8f;

// ---------------------------------------------------------------------------
// K1: flat[b][o] = read_inputs[b] . W[o] + bias[o]      (one wave per output)
// ---------------------------------------------------------------------------
__global__ __launch_bounds__(32) void k1_proj(const float* __restrict__ x,
                                              const float* __restrict__ W,
                                              const float* __restrict__ bias,
                                              float* __restrict__ flat) {
    const int o    = blockIdx.x;
    const int bb   = blockIdx.y;
    const int lane = threadIdx.x;
    const float* xr = x + (size_t)bb * DIN_;
    const float* wr = W + (size_t)o * DIN_;
    float s = 0.f;
    for (int j = lane; j < DIN_; j += 32) s += xr[j] * wr[j];
    #pragma unroll
    for (int off = 16; off > 0; off >>= 1) s += __shfl_xor(s, off, 32);
    if (lane == 0) flat[bb * OUTD_ + o] = s + bias[o];
}

// ---------------------------------------------------------------------------
// K1b: normalize read_keys, extract read_strengths     (one wave per (b,h))
// ---------------------------------------------------------------------------
__global__ __launch_bounds__(32) void k1b_keys(const float* __restrict__ flat,
                                               float* __restrict__ keys,
                                               float* __restrict__ strengths) {
    const int bh = blockIdx.x;
    const int bb = bh >> 3, h = bh & 7;
    const int lane = threadIdx.x;
    const float* src = flat + bb * OUTD_ + h * M_;
    float a0 = src[lane];
    float a1 = src[lane + 32];
    float ss = a0 * a0 + a1 * a1;
    #pragma unroll
    for (int off = 16; off > 0; off >>= 1) ss += __shfl_xor(ss, off, 32);
    float inv = 1.f / fmaxf(sqrtf(ss), 1e-12f);
    keys[(size_t)bh * M_ + lane]      = a0 * inv;
    keys[(size_t)bh * M_ + lane + 32] = a1 * inv;
    if (lane == 0) strengths[bh] = flat[bb * OUTD_ + DIN_ + h];
}

// ---------------------------------------------------------------------------
// K2: weighted cosine scores via V_WMMA_F32_16X16X4_F32.
//   C(16r x 16h) = memTile(16r x 64m) x keysT(64m x 16h), K split into 16x4.
//   scores[b][r][h] = strength[h] * dot / ||mem_row||   (h < 8 stored)
// grid = (32 chunks, 16 batches), 256 threads = 8 waves, 256 rows per wave.
// ---------------------------------------------------------------------------
__global__ __launch_bounds__(256) void k2_scores(const float* __restrict__ mem,
                                                 const float* __restrict__ keys,
                                                 const float* __restrict__ strengths,
                                                 float* __restrict__ scores) {
    const int bb    = blockIdx.y;
    const int chunk = blockIdx.x;            // 32 chunks of 2048 rows
    const int wave  = threadIdx.x >> 5;
    const int lane  = threadIdx.x & 31;
    const int h     = lane & 15;             // column of C (head id)
    const int grp   = lane >> 4;             // 0 / 1 (half-wave)

    const float sh = (h < H_) ? strengths[bb * H_ + h] : 0.f;

    // Preload B fragments: bf[kk][v] = keys[b][h][4*kk + v + 2*grp] (zero for h>=8)
    v2f bf[16];
    #pragma unroll
    for (int kk = 0; kk < 16; ++kk) {
        if (h < H_) {
            bf[kk] = *(const v2f*)(keys + ((size_t)bb * H_ + h) * M_ + 4 * kk + 2 * grp);
        } else {
            v2f z; z.x = 0.f; z.y = 0.f;
            bf[kk] = z;
        }
    }

    const int rowBase = chunk * 2048 + wave * 256;
    const float* memB = mem + (size_t)bb * R_ * M_;

    for (int t = 0; t < 16; ++t) {
        const int tb = rowBase + t * 16;
        // lane l owns row tb + (l%16); K offset v + 2*grp within each 4-chunk
        const float* rowPtr = memB + (size_t)(tb + h) * M_ + 2 * grp;

        if (t < 15) __builtin_prefetch(rowPtr + 16 * M_, 0, 1);  // next tile -> global_prefetch

        v8f acc = {};
        float ss = 0.f;
        #pragma unroll
        for (int kk = 0; kk < 16; ++kk) {
            v2f a = *(const v2f*)(rowPtr + 4 * kk);   // A frag: mem[row][4kk + v + 2*grp]
            ss += a.x * a.x + a.y * a.y;              // free row-norm partial
            acc = __builtin_amdgcn_wmma_f32_16x16x4_f32(
                      false, a, false, bf[kk], (short)0, acc, false, false);
        }
        // combine the two half-row sumsq partials (lanes l and l^16 share a row)
        ss += __shfl_xor(ss, 16, 32);
        const float inv = 1.f / fmaxf(sqrtf(ss), 1e-12f);

        // broadcast inv-norms to score owners (ALL lanes participate: sources 0..15 active)
        float invr[8];
        #pragma unroll
        for (int d = 0; d < 8; ++d) invr[d] = __shfl(inv, d + 8 * grp, 32);

        if (h < H_) {
            #pragma unroll
            for (int d = 0; d < 8; ++d) {
                const int r = tb + d + 8 * grp;       // C VGPR d -> row d (+8 for upper half)
                scores[((size_t)bb * R_ + r) * 8 + h] = acc[d] * invr[d] * sh;
            }
        }
    }
}

// ---------------------------------------------------------------------------
// K3: per (b,h): top-32 of 65536 scores, softmax, gather + weighted sum.
// 128 threads; per-thread register top-32 (unrolled insertion), LDS merge
// with 32 argmax-extraction rounds (descending order == jax.lax.top_k).
// ---------------------------------------------------------------------------
#define NT3 128
__global__ __launch_bounds__(NT3) void k3_topk(const float* __restrict__ scores,
                                               const float* __restrict__ mem,
                                               const float* __restrict__ strengths,
                                               float* __restrict__ out) {
    const int bh = blockIdx.x;               // 128 = B*H
    const int bb = bh >> 3, h = bh & 7;
    const int t  = threadIdx.x;

    __shared__ float sv[NT3 * K_];
    __shared__ int   si[NT3 * K_];
    __shared__ float rv[NT3];
    __shared__ int   rp[NT3];
    __shared__ float topv[K_];
    __shared__ int   topi[K_];
    __shared__ float wts[K_];

    // per-thread top-32 (stays in VGPRs: static indexing only)
    float lv[K_]; int li[K_];
    #pragma unroll
    for (int k = 0; k < K_; ++k) { lv[k] = -FLT_MAX; li[k] = 0; }

    const float* sb = scores + (size_t)bb * R_ * 8 + h;
    for (int j = 0; j < R_ / NT3; ++j) {
        const int r = t + NT3 * j;
        const float v = sb[(size_t)r * 8];
        if (v > lv[K_ - 1]) {
            lv[K_ - 1] = v; li[K_ - 1] = r;
            #pragma unroll
            for (int k = K_ - 1; k > 0; --k) {
                if (lv[k] > lv[k - 1]) {
                    float tv = lv[k]; lv[k] = lv[k - 1]; lv[k - 1] = tv;
                    int   ti = li[k]; li[k] = li[k - 1]; li[k - 1] = ti;
                }
            }
        }
    }
    #pragma unroll
    for (int k = 0; k < K_; ++k) { sv[t * K_ + k] = lv[k]; si[t * K_ + k] = li[k]; }
    __syncthreads();

    // 32 extraction rounds over 4096 candidates
    for (int k = 0; k < K_; ++k) {
        float best = -FLT_MAX; int bp = 0;
        for (int j = t; j < NT3 * K_; j += NT3) {
            const float v = sv[j];
            if (v > best) { best = v; bp = j; }
        }
        rv[t] = best; rp[t] = bp;
        __syncthreads();
        for (int s = NT3 / 2; s > 0; s >>= 1) {
            if (t < s && rv[t + s] > rv[t]) { rv[t] = rv[t + s]; rp[t] = rp[t + s]; }
            __syncthreads();
        }
        if (t == 0) {
            topv[k] = rv[0];
            topi[k] = si[rp[0]];
            sv[rp[0]] = -FLT_MAX;
        }
        __syncthreads();
    }

    // softmax over the 32 selected weighted scores (wave 0)
    if (t < K_) {
        const float e = __expf(topv[t] - topv[0]);
        float ssum = e;
        #pragma unroll
        for (int off = 16; off > 0; off >>= 1) ssum += __shfl_xor(ssum, off, 32);
        wts[t] = e / ssum;
    }
    __syncthreads();

    // memory_reads[b][h][m] = sum_k w_k * mem[b][idx_k][m]
    if (t < M_) {
        const float* memB = mem + (size_t)bb * R_ * M_;
        float acc = 0.f;
        for (int k = 0; k < K_; ++k) acc += wts[k] * memB[(size_t)topi[k] * M_ + t];
        out[bb * (H_ * M_) + h * M_ + t] = acc;   // flat_reads
    }
    if (t < K_) {
        out[8192  + bh * K_ + t] = wts[t];        // read_weights
        out[12288 + bh * K_ + t] = (float)topi[t];// read_indices (exact as float)
    }
    if (t == 0) out[16384 + bh] = strengths[bh];  // read_strengths
}

// ---------------------------------------------------------------------------
extern "C" void kernel_launch(void* const* d_in, const int* in_sizes, int n_in,
                              void* d_out, int out_size, void* d_ws, size_t ws_size,
                              hipStream_t stream) {
    const float* x      = (const float*)d_in[0];  // read_inputs (B, D_IN)
    const float* memst  = (const float*)d_in[1];  // mem_state   (B, R, M)
    const float* W      = (const float*)d_in[2];  // W           (OUT_DIM, D_IN)
    const float* bias   = (const float*)d_in[3];  // b           (OUT_DIM,)
    float* out = (float*)d_out;
    float* ws  = (float*)d_ws;

    float* flat      = ws;                        // 16*520      = 8320
    float* keys      = ws + 8320;                 // 16*8*64     = 8192
    float* strengths = ws + 16512;                // 128
    float* scores    = ws + 16640;                // 16*65536*8  = 8388608 (32 MB, L2-resident)

    k1_proj  <<<dim3(OUTD_, B_), 32,  0, stream>>>(x, W, bias, flat);
    k1b_keys <<<dim3(B_ * H_),   32,  0, stream>>>(flat, keys, strengths);
    k2_scores<<<dim3(32, B_),    256, 0, stream>>>(memst, keys, strengths, scores);
    k3_topk  <<<dim3(B_ * H_),   NT3, 0, stream>>>(scores, memst, strengths, out);
}